// GraphConv_23716809409087
// MI455X (gfx1250) — compile-verified
//
#include <hip/hip_runtime.h>
#include <hip/hip_bf16.h>

#define IN_DIM  128
#define OUT_DIM 128

typedef __attribute__((ext_vector_type(2))) float v2f;
typedef __attribute__((ext_vector_type(8))) float v8f;

// ---------------------------------------------------------------------------
// Kernel 1: support = X @ W via V_WMMA_F32_16X16X4_F32.
// One wave computes a 16x128 band. Block = 128 threads = 4 waves = 64 rows.
// W is staged in LDS with K-pairs interleaved so each B fragment is a single
// aligned ds_load_b64: lw[(k/2)*256 + col*2 + (k&1)].
// ---------------------------------------------------------------------------
__global__ __launch_bounds__(128) void gemm_xw_wmma(
    const float* __restrict__ X,   // [n_nodes, IN_DIM]
    const float* __restrict__ W,   // [IN_DIM, OUT_DIM]
    float* __restrict__ S,         // [n_nodes, OUT_DIM]
    int n_nodes)
{
    __shared__ float lw[IN_DIM * OUT_DIM]; // 64 KB, pair-interleaved layout

    const int tid = threadIdx.x;

    // Cooperative staging: read W as float4, scatter into pair-interleaved LDS.
    {
        const float4* __restrict__ W4 = (const float4*)W;
        for (int i = tid; i < (IN_DIM * OUT_DIM) / 4; i += 128) {
            float4 w = W4[i];
            const int k   = (i * 4) >> 7;      // source row
            const int col = (i * 4) & 127;     // source col (multiple of 4)
            float* dst = &lw[(k >> 1) * 256 + col * 2 + (k & 1)];
            dst[0] = w.x; dst[2] = w.y; dst[4] = w.z; dst[6] = w.w;
        }
    }
    __syncthreads();

    const int wave  = tid >> 5;
    const int lane  = tid & 31;
    const int mrow  = lane & 15;         // M (A) / N (B,C) within tile
    const int khalf = lane >> 4;         // 0: K pair {k,k+1}, 1: {k+2,k+3}

    const int row0 = (blockIdx.x * 4 + wave) * 16;
    int grow = row0 + mrow;
    if (grow >= n_nodes) grow = n_nodes - 1;          // clamp A loads (tail)
    const float* __restrict__ xrow = X + (size_t)grow * IN_DIM;

    v8f acc[8] = {};

    for (int k = 0; k < IN_DIM; k += 4) {
        const int ka = k + 2 * khalf;
        // A fragment: one aligned 8-byte global load
        const v2f a = *(const v2f*)(xrow + ka);
        // B fragments: 8 aligned ds_load_b64 at constant offsets
        const float* bp = &lw[(ka >> 1) * 256 + mrow * 2];
        #pragma unroll
        for (int t = 0; t < 8; ++t) {
            const v2f b = *(const v2f*)(bp + t * 32);
            acc[t] = __builtin_amdgcn_wmma_f32_16x16x4_f32(
                false, a, false, b, (short)0, acc[t], false, false);
        }
    }

    // C/D layout: VGPR j holds M=j (lanes 0-15) / M=j+8 (lanes 16-31), N=lane%16
    const int rbase = row0 + 8 * khalf;
    float* __restrict__ sp = S + (size_t)rbase * OUT_DIM + mrow;
    if (row0 + 16 <= n_nodes) {
        // wave-uniform fast path: 64 unguarded stores, constant offsets
        #pragma unroll
        for (int j = 0; j < 8; ++j)
            #pragma unroll
            for (int t = 0; t < 8; ++t)
                sp[j * OUT_DIM + t * 16] = acc[t][j];
    } else {
        #pragma unroll
        for (int j = 0; j < 8; ++j)
            if (rbase + j < n_nodes)
                #pragma unroll
                for (int t = 0; t < 8; ++t)
                    sp[j * OUT_DIM + t * 16] = acc[t][j];
    }
}

// ---------------------------------------------------------------------------
// Kernel 2: out[n][d] = bias[d]  (float4 writes; d_out is poisoned)
// ---------------------------------------------------------------------------
__global__ __launch_bounds__(256) void init_out(
    float4* __restrict__ out4, const float* __restrict__ bias, int total4)
{
    const int i = blockIdx.x * blockDim.x + threadIdx.x;
    if (i < total4) {
        const int d = (i & (OUT_DIM / 4 - 1)) * 4;
        out4[i] = *(const float4*)(bias + d);
    }
}

// ---------------------------------------------------------------------------
// Kernel 3: SpMM scatter, one wave per edge. Edge index forced wave-uniform
// (readfirstlane) so row/col/val become scalar loads; gather is one
// global_load_b128 per lane; scatter is 4 no-return global_atomic_add_f32.
// ---------------------------------------------------------------------------
__global__ __launch_bounds__(256) void scatter_edges(
    const float* __restrict__ S,
    const int*   __restrict__ erow,
    const int*   __restrict__ ecol,
    const float* __restrict__ eval,
    float*       __restrict__ out,
    int n_edges)
{
    const int wid  = blockIdx.x * 8 + (threadIdx.x >> 5);   // 8 waves / block
    const int lane = threadIdx.x & 31;
    const int e    = __builtin_amdgcn_readfirstlane(wid);
    if (e >= n_edges) return;

    const int   r = erow[e];
    const int   c = ecol[e];
    const float v = eval[e];

    const float4 s = ((const float4*)(S + (size_t)c * OUT_DIM))[lane];
    float* op = out + (size_t)r * OUT_DIM + lane * 4;

    atomicAdd(op + 0, v * s.x);
    atomicAdd(op + 1, v * s.y);
    atomicAdd(op + 2, v * s.z);
    atomicAdd(op + 3, v * s.w);
}

// ---------------------------------------------------------------------------
extern "C" void kernel_launch(void* const* d_in, const int* in_sizes, int n_in,
                              void* d_out, int out_size, void* d_ws, size_t ws_size,
                              hipStream_t stream)
{
    const float* X    = (const float*)d_in[0];   // fea_input [N, 128]
    const int*   erow = (const int*)  d_in[1];   // edge_row  [E]
    const int*   ecol = (const int*)  d_in[2];   // edge_col  [E]
    const float* eval = (const float*)d_in[3];   // edge_val  [E]
    const float* W    = (const float*)d_in[4];   // weight [128, 128]
    const float* bias = (const float*)d_in[5];   // bias [128]

    float* out = (float*)d_out;
    float* S   = (float*)d_ws;                   // support [N, 128] = 51.2 MB

    const int n_nodes = in_sizes[0] / IN_DIM;
    const int n_edges = in_sizes[1];

    // 1) support = X @ W  (WMMA fp32)
    {
        const int rows_per_block = 64;           // 4 waves x 16 rows
        const int grid = (n_nodes + rows_per_block - 1) / rows_per_block;
        gemm_xw_wmma<<<grid, 128, 0, stream>>>(X, W, S, n_nodes);
    }

    // 2) out = broadcast(bias)
    {
        const int total4 = n_nodes * OUT_DIM / 4;
        const int grid   = (total4 + 255) / 256;
        init_out<<<grid, 256, 0, stream>>>((float4*)out, bias, total4);
    }

    // 3) out[row] += val * support[col]  (atomic scatter, 1 wave / edge)
    {
        const int grid = (n_edges + 7) / 8;      // 8 edges (waves) per block
        scatter_edges<<<grid, 256, 0, stream>>>(S, erow, ecol, eval, out, n_edges);
    }
}